// WrappedBERT_55774445306483
// MI455X (gfx1250) — compile-verified
//
#include <hip/hip_runtime.h>

// Problem constants from the reference
constexpr int BATCH = 512;
constexpr int SEQ   = 512;
constexpr int DIM   = 768;
constexpr int CLS_ID = 101, SEP_ID = 102, PAD_ID = 0;

constexpr int TILES_PER_WAVE = (DIM / 16) / 8;  // 48 d-tiles / 8 waves = 6

typedef __attribute__((ext_vector_type(2))) float v2f;
typedef __attribute__((ext_vector_type(8))) float v8f;

__launch_bounds__(256)
__global__ void masked_mean_wmma(const float* __restrict__ embed,
                                 const int*   __restrict__ ids,
                                 float*       __restrict__ out) {
    __shared__ float smask[SEQ];
    __shared__ float scount;

    const int b    = blockIdx.x;
    const int tid  = threadIdx.x;
    const int wave = tid >> 5;   // 0..7
    const int lane = tid & 31;

    if (tid == 0) scount = 0.0f;
    __syncthreads();

    // ---- Phase 1: build mask in LDS, count valid tokens (uniform control flow) ----
    float local = 0.0f;
    #pragma unroll
    for (int s = tid; s < SEQ; s += 256) {
        const int id = ids[(size_t)b * SEQ + s];
        const float m = (id != PAD_ID && id != CLS_ID && id != SEP_ID) ? 1.0f : 0.0f;
        smask[s] = m;
        local += m;
    }
    atomicAdd(&scount, local);           // ds_add_f32
    __syncthreads();
    const float inv = 1.0f / scount;

    // ---- Phase 2: WMMA reduction over S ----
    // B-matrix (4x16 f32) layout: lane = N (d-column), lanes 0-15 hold K=0/1 in
    // the two VGPRs, lanes 16-31 hold K=2/3. A-matrix rows are all identical
    // (mask broadcast), so every C row accumulates the masked column sums.
    const int col = lane & 15;            // d column within tile
    const int kb  = (lane >> 4) << 1;     // 0 for lanes 0-15, 2 for lanes 16-31
    const float* eb = embed + (size_t)b * SEQ * DIM;

    v8f acc[TILES_PER_WAVE];
    #pragma unroll
    for (int t = 0; t < TILES_PER_WAVE; ++t)
        acc[t] = (v8f){0.f, 0.f, 0.f, 0.f, 0.f, 0.f, 0.f, 0.f};

    for (int sc = 0; sc < SEQ; sc += 4) {
        v2f a;                            // A rows broadcast: {m[sc+kb], m[sc+kb+1]}
        a.x = smask[sc + kb];
        a.y = smask[sc + kb + 1];

        const float* r0 = eb + (size_t)(sc + kb) * DIM + col;
        const float* r1 = r0 + DIM;

        if (sc + 16 < SEQ)                // stream ahead: global_prefetch_b8
            __builtin_prefetch(r0 + (size_t)16 * DIM + (wave << 4), 0, 0);

        #pragma unroll
        for (int t = 0; t < TILES_PER_WAVE; ++t) {
            const int d0 = (wave + (t << 3)) << 4;   // 16*(wave + 8t)
            v2f bv;
            bv.x = r0[d0];                // K = kb   row of this 4x16 tile
            bv.y = r1[d0];                // K = kb+1 row
            acc[t] = __builtin_amdgcn_wmma_f32_16x16x4_f32(
                         /*neg_a=*/false, a, /*neg_b=*/false, bv,
                         /*c_mod=*/(short)0, acc[t],
                         /*reuse_a=*/false, /*reuse_b=*/false);
        }
    }

    // ---- Phase 3: row M=0 of C (VGPR0, lanes 0-15) holds the column sums ----
    if (lane < 16) {
        float* ob = out + (size_t)b * DIM + col;
        #pragma unroll
        for (int t = 0; t < TILES_PER_WAVE; ++t) {
            const int d0 = (wave + (t << 3)) << 4;
            ob[d0] = acc[t][0] * inv;
        }
    }
}

extern "C" void kernel_launch(void* const* d_in, const int* in_sizes, int n_in,
                              void* d_out, int out_size, void* d_ws, size_t ws_size,
                              hipStream_t stream) {
    (void)in_sizes; (void)n_in; (void)out_size; (void)d_ws; (void)ws_size;
    const float* embed = (const float*)d_in[0];
    const int*   ids   = (const int*)d_in[1];
    float*       out   = (float*)d_out;
    hipLaunchKernelGGL(masked_mean_wmma, dim3(BATCH), dim3(256), 0, stream,
                       embed, ids, out);
}